// ScaledDotProductAttention_6262062318235
// MI455X (gfx1250) — compile-verified
//
#include <hip/hip_runtime.h>

typedef __attribute__((ext_vector_type(2))) float v2f;
typedef __attribute__((ext_vector_type(4))) float v4f;
typedef __attribute__((ext_vector_type(8))) float v8f;

#define LSEQ  2048        // sequence length (Lq == Lk)
#define DH    64          // head dim
#define SSTR  2052        // padded LDS row stride (floats): 8*2052 % 64 == 32 -> no bank conflict between lane halves
#define NWAVE 8           // waves per workgroup (wave32 -> 256 threads)
#define WSL   (LSEQ / NWAVE)   // 256 k-columns per wave slice
#define NTILE (WSL / 16)       // 16 k-tiles of 16 columns per wave
#define QT    16          // q rows per workgroup

__global__ __launch_bounds__(256, 2)
void attn_fused(const float* __restrict__ gq, const float* __restrict__ gk,
                const float* __restrict__ gv, const float* __restrict__ gb,
                float* __restrict__ gout, float* __restrict__ gw)
{
    // ~133.5 KB static LDS: CDNA5 WGP has 320 KB -> 2 workgroups / WGP
    __shared__ float S[QT * SSTR];        // raw scores, then normalized weights
    __shared__ float redM[NWAVE * 16];    // per-wave partial row max
    __shared__ float redS[NWAVE * 16];    // per-wave partial exp-sum
    __shared__ float finM[16];            // final row max
    __shared__ float finI[16];            // final 1/rowsum
    __shared__ float Oacc[QT * DH];       // cross-wave O accumulator

    const int tid  = threadIdx.x;
    const int wid  = tid >> 5;
    const int lane = tid & 31;
    const int bh   = blockIdx.x >> 7;          // b*H + h
    const int q0   = (blockIdx.x & 127) << 4;  // q-tile start row

    const size_t rb = (size_t)bh * LSEQ;
    const float* qp = gq + (rb + (size_t)q0) * DH;
    const float* kp = gk + rb * DH;
    const float* vp = gv + rb * DH;
    const float* bp = gb + ((size_t)bh * LSEQ + q0) * (size_t)LSEQ;
    float*       wp = gw + ((size_t)bh * LSEQ + q0) * (size_t)LSEQ;

    const int m16   = lane & 15;               // A: M row / B: N col / D: N col
    const int koff  = (lane < 16) ? 0 : 2;     // K sub-index for lane half
    const int mhalf = (lane < 16) ? 0 : 8;     // D: M offset for lane half

    // ---------------- load scaled Q fragment (A operand, 16 chunks of K=4)
    v2f qa[16];
    {
        const float* qr = qp + (size_t)m16 * DH + koff;
        #pragma unroll
        for (int c = 0; c < 16; ++c) {
            qa[c].x = qr[4 * c]     * 0.125f;   // fold 1/sqrt(64) into Q
            qa[c].y = qr[4 * c + 1] * 0.125f;
        }
    }

    // ---------------- phase 1: S = (Q*scale)K^T + bias  ->  LDS
    for (int t = 0; t < NTILE; ++t) {
        const int colb = (wid * NTILE + t) * 16;
        const float* kr = kp + (size_t)(colb + m16) * DH + koff;  // B: lane = N col
        v8f acc = {0.f, 0.f, 0.f, 0.f, 0.f, 0.f, 0.f, 0.f};
        #pragma unroll
        for (int c = 0; c < 16; ++c) {
            v2f b;
            b.x = kr[4 * c];
            b.y = kr[4 * c + 1];
            acc = __builtin_amdgcn_wmma_f32_16x16x4_f32(false, qa[c], false, b,
                                                        (short)0, acc, false, false);
        }
        const int col = colb + m16;
        #pragma unroll
        for (int r = 0; r < 8; ++r) {
            const int m = r + mhalf;
            S[m * SSTR + col] = acc[r] + bp[(size_t)m * LSEQ + col];
        }
    }
    __syncthreads();

    // ---------------- phase 2a: row max (per-wave partial, then combine)
    {
        const int c0 = wid * WSL + (lane >> 4) * (WSL / 2);
        float mx = -3.402823466e+38f;
        for (int c = 0; c < WSL / 2; ++c)
            mx = fmaxf(mx, S[m16 * SSTR + c0 + c]);
        mx = fmaxf(mx, __shfl_xor(mx, 16, 32));
        if (lane < 16) redM[wid * 16 + lane] = mx;
    }
    __syncthreads();
    if (tid < 16) {
        float mx = redM[tid];
        for (int i = 1; i < NWAVE; ++i) mx = fmaxf(mx, redM[i * 16 + tid]);
        finM[tid] = mx;
    }
    for (int i = tid; i < QT * DH; i += 256) Oacc[i] = 0.f;  // zero O accumulator
    __syncthreads();

    // ---------------- phase 2b: exp-sum
    {
        const int c0 = wid * WSL + (lane >> 4) * (WSL / 2);
        const float fm = finM[m16];
        float sm = 0.f;
        for (int c = 0; c < WSL / 2; ++c)
            sm += __expf(S[m16 * SSTR + c0 + c] - fm);
        sm += __shfl_xor(sm, 16, 32);
        if (lane < 16) redS[wid * 16 + lane] = sm;
    }
    __syncthreads();
    if (tid < 16) {
        float sm = 0.f;
        for (int i = 0; i < NWAVE; ++i) sm += redS[i * 16 + tid];
        finI[tid] = 1.0f / sm;
    }
    __syncthreads();

    // ---------------- phase 3: normalize slice, write weights, rewrite LDS
    {
        const int cbase = wid * WSL + lane * 4;
        #pragma unroll
        for (int m = 0; m < 16; ++m) {
            const float fm = finM[m];
            const float fi = finI[m];
            #pragma unroll
            for (int c = 0; c < 2; ++c) {
                const int col = cbase + c * 128;
                v4f sv = *(const v4f*)&S[m * SSTR + col];
                v4f pv;
                pv.x = __expf(sv.x - fm) * fi;
                pv.y = __expf(sv.y - fm) * fi;
                pv.z = __expf(sv.z - fm) * fi;
                pv.w = __expf(sv.w - fm) * fi;
                *(v4f*)&S[m * SSTR + col] = pv;                     // for P*V
                *(v4f*)&wp[(size_t)m * LSEQ + col] = pv;            // weights out
            }
        }
    }
    // no barrier needed: each wave consumes only its own slice of S below

    // ---------------- phase 4: O_partial = P_slice * V_slice   (WMMA, K=4 chunks)
    {
        v8f accO[4];
        #pragma unroll
        for (int nt = 0; nt < 4; ++nt)
            accO[nt] = (v8f){0.f, 0.f, 0.f, 0.f, 0.f, 0.f, 0.f, 0.f};

        const int kb0 = wid * WSL;
        for (int c = 0; c < WSL / 4; ++c) {
            const int kb = kb0 + 4 * c;
            v2f a = *(const v2f*)&S[m16 * SSTR + kb + koff];        // A: lane = M
            const float* vr = vp + (size_t)(kb + koff) * DH + m16;  // B: lane = N
            #pragma unroll
            for (int nt = 0; nt < 4; ++nt) {
                v2f b;
                b.x = vr[nt * 16];
                b.y = vr[nt * 16 + DH];
                accO[nt] = __builtin_amdgcn_wmma_f32_16x16x4_f32(false, a, false, b,
                                                                 (short)0, accO[nt],
                                                                 false, false);
            }
        }
        #pragma unroll
        for (int nt = 0; nt < 4; ++nt) {
            #pragma unroll
            for (int r = 0; r < 8; ++r) {
                const int m = r + mhalf;
                atomicAdd(&Oacc[m * DH + nt * 16 + m16], accO[nt][r]);  // ds_add_f32
            }
        }
    }
    __syncthreads();

    // ---------------- write O: 1024 floats = 256 float4, one per thread
    {
        const int m  = tid >> 4;
        const int n4 = (tid & 15) * 4;
        v4f ov = *(const v4f*)&Oacc[m * DH + n4];
        *(v4f*)&gout[((size_t)bh * LSEQ + q0 + m) * DH + n4] = ov;
    }
}

extern "C" void kernel_launch(void* const* d_in, const int* in_sizes, int n_in,
                              void* d_out, int out_size, void* d_ws, size_t ws_size,
                              hipStream_t stream) {
    (void)in_sizes; (void)n_in; (void)out_size; (void)d_ws; (void)ws_size;
    const float* q = (const float*)d_in[0];
    const float* k = (const float*)d_in[1];
    const float* v = (const float*)d_in[2];
    const float* b = (const float*)d_in[3];
    float* out = (float*)d_out;                                 // [B,H,L,D] first
    float* w   = out + (size_t)2 * 16 * 2048 * 64;              // then [B,H,L,L]
    const int blocks = 2 * 16 * (LSEQ / QT);                    // 4096
    attn_fused<<<dim3(blocks), dim3(256), 0, stream>>>(q, k, v, b, out, w);
}